// PVRNet_69303592288277
// MI455X (gfx1250) — compile-verified
//
#include <hip/hip_runtime.h>
#include <math.h>

#define BN_INV 0.9999950000374998f   // 1/sqrt(1+1e-5)
#define NPTS 1024
#define KNN_K 20

typedef float v2f __attribute__((ext_vector_type(2)));
typedef float v8f __attribute__((ext_vector_type(8)));

__device__ __forceinline__ float act_apply(float y, int act, float slope) {
  if (act == 1) return fmaxf(y, 0.f);
  if (act == 2) return y > 0.f ? y : y * slope;
  if (act == 3) return tanhf(y);
  if (act == 4) return 1.f / (1.f + expf(-y));
  return y;
}

// ---------------------------------------------------------------------------
// Generic batch-16 GEMM: out(16,O) = A(16,K) @ W(O,K)^T + bias, then optional
// BN (g*INV,beta per output col) and activation. One wave per 16x16 tile,
// V_WMMA_F32_16X16X4_F32 over K. Weight-streaming bound by design.
// ---------------------------------------------------------------------------
__global__ __launch_bounds__(32)
void gemm16_wmma(const float* __restrict__ A, const float* __restrict__ W,
                 const float* __restrict__ bias,
                 const float* __restrict__ bng, const float* __restrict__ bnb,
                 float* __restrict__ out, int K, int O, int act, float slope) {
  const int lane = threadIdx.x & 31;
  const int half = lane >> 4;            // selects K sub-pair per ISA A/B layout
  const int l    = lane & 15;            // A row (M) / output column (N)
  const int o0   = blockIdx.x * 16;
  const int ocol = o0 + l;
  const int oc_c = ocol < O ? ocol : (O - 1);
  const float wmask = (ocol < O) ? 1.f : 0.f;
  const float* __restrict__ arow = A + (size_t)l * K;
  const float* __restrict__ wrow = W + (size_t)oc_c * K;

  v8f acc = {};
  for (int kk = 0; kk < K; kk += 4) {
    const int ka = kk + 2 * half;
    v2f a, b;
    a.x = arow[ka];         a.y = arow[ka + 1];
    b.x = wrow[ka] * wmask; b.y = wrow[ka + 1] * wmask;
    acc = __builtin_amdgcn_wmma_f32_16x16x4_f32(false, a, false, b,
                                                (short)0, acc, false, false);
  }
  if (ocol >= O) return;
  const float bi = bias ? bias[ocol] : 0.f;
  const float gg = bng ? bng[ocol] * BN_INV : 1.f;
  const float bb = bnb ? bnb[ocol] : 0.f;
#pragma unroll
  for (int r = 0; r < 8; ++r) {
    const int mrow = half * 8 + r;       // D layout: lanes<16 hold M=0..7, else 8..15
    float y = acc[r] + bi;
    y = y * gg + bb;
    out[(size_t)mrow * O + ocol] = act_apply(y, act, slope);
  }
}

// ---------------------------------------------------------------------------
// out(B,O) = max_n relu(bn(W(O,C) @ X(B,C,N) + bias)).  Wave per (b, 16-o tile),
// WMMA over C per 16-column tile of N, running max in registers, final
// cross-lane max via shfl_xor within each 16-lane half. O,N mult of 16; C of 4.
// ---------------------------------------------------------------------------
__global__ __launch_bounds__(32)
void convmax_wmma(const float* __restrict__ X, const float* __restrict__ W,
                  const float* __restrict__ bias, const float* __restrict__ bng,
                  const float* __restrict__ bnb, float* __restrict__ out,
                  int C, int N, int O, int ldb) {
  const int b = blockIdx.y;
  const int o0 = blockIdx.x * 16;
  const int lane = threadIdx.x & 31;
  const int half = lane >> 4;
  const int l = lane & 15;
  const float* __restrict__ x = X + (size_t)b * ldb;
  const float* __restrict__ wrow = W + (size_t)(o0 + l) * C;

  float bi[8], gg[8], bb[8];
#pragma unroll
  for (int r = 0; r < 8; ++r) {
    const int oc = o0 + half * 8 + r;
    bi[r] = bias[oc]; gg[r] = bng[oc] * BN_INV; bb[r] = bnb[oc];
  }
  v8f vmax;
#pragma unroll
  for (int r = 0; r < 8; ++r) vmax[r] = -3.0e38f;

  for (int n0 = 0; n0 < N; n0 += 16) {
    v8f acc = {};
    for (int kk = 0; kk < C; kk += 4) {
      const int ka = kk + 2 * half;
      v2f a, bvec;
      a.x = wrow[ka]; a.y = wrow[ka + 1];
      bvec.x = x[(size_t)ka * N + n0 + l];
      bvec.y = x[(size_t)(ka + 1) * N + n0 + l];
      acc = __builtin_amdgcn_wmma_f32_16x16x4_f32(false, a, false, bvec,
                                                  (short)0, acc, false, false);
    }
#pragma unroll
    for (int r = 0; r < 8; ++r) {
      float y = fmaxf((acc[r] + bi[r]) * gg[r] + bb[r], 0.f);
      vmax[r] = fmaxf(vmax[r], y);
    }
  }
#pragma unroll
  for (int r = 0; r < 8; ++r) {
    float v = vmax[r];
    for (int mask = 1; mask < 16; mask <<= 1)
      v = fmaxf(v, __shfl_xor(v, mask, 32));
    vmax[r] = v;
  }
  if (l == 0) {
#pragma unroll
    for (int r = 0; r < 8; ++r)
      out[(size_t)b * O + o0 + half * 8 + r] = vmax[r];
  }
}

// ---------------------------------------------------------------------------
// KNN: per (b,n) block, -||x_n - x_m||^2 for all m in LDS, then 20x argmax.
// ---------------------------------------------------------------------------
__global__ __launch_bounds__(256)
void knn_kernel(const float* __restrict__ X, int* __restrict__ idx, int C, int ldb) {
  __shared__ float ctr[64];
  __shared__ float dist[NPTS];
  __shared__ float rv[256];
  __shared__ int ri[256];
  const int b = blockIdx.y, n = blockIdx.x, t = threadIdx.x;
  const float* __restrict__ x = X + (size_t)b * ldb;
  for (int c = t; c < C; c += 256) ctr[c] = x[(size_t)c * NPTS + n];
  __syncthreads();
  for (int m = t; m < NPTS; m += 256) {
    float d = 0.f;
    for (int c = 0; c < C; ++c) {
      float df = x[(size_t)c * NPTS + m] - ctr[c];
      d += df * df;
    }
    dist[m] = -d;
  }
  __syncthreads();
  for (int k = 0; k < KNN_K; ++k) {
    float bv = -3.0e38f; int bidx = 0;
    for (int m = t; m < NPTS; m += 256) {
      float v = dist[m];
      if (v > bv) { bv = v; bidx = m; }
    }
    rv[t] = bv; ri[t] = bidx;
    __syncthreads();
    for (int s = 128; s > 0; s >>= 1) {
      if (t < s) {
        float v = rv[t + s]; int i2 = ri[t + s];
        if (v > rv[t] || (v == rv[t] && i2 < ri[t])) { rv[t] = v; ri[t] = i2; }
      }
      __syncthreads();
    }
    if (t == 0) {
      idx[((size_t)b * NPTS + n) * KNN_K + k] = ri[0];
      dist[ri[0]] = -3.4e38f;
    }
    __syncthreads();
  }
}

// ---------------------------------------------------------------------------
// Fused edge-conv + max over k: out[b,o,n] = max_k relu(bn(W[o,:2C]·e + bias))
// e = [nbr-ctr, ctr]. One block per (b,n), O threads. Writes into a channel
// slice of the concat buffer (ldo = total channel stride of that buffer).
// ---------------------------------------------------------------------------
__global__ void edgeconv_max(const float* __restrict__ X, const int* __restrict__ idx,
                             const float* __restrict__ W, const float* __restrict__ bias,
                             const float* __restrict__ bng, const float* __restrict__ bnb,
                             float* __restrict__ out, int C, int O, int ldbx, int ldo) {
  __shared__ float ctr[64];
  __shared__ float nbr[64];
  const int b = blockIdx.y, n = blockIdx.x, t = threadIdx.x;
  const float* __restrict__ x = X + (size_t)b * ldbx;
  for (int c = t; c < C; c += blockDim.x) ctr[c] = x[(size_t)c * NPTS + n];
  const float bi = bias[t];
  const float gg = bng[t] * BN_INV;
  const float bb = bnb[t];
  const float* __restrict__ w0 = W + (size_t)t * 2 * C;
  float vmax = -3.0e38f;
  for (int k = 0; k < KNN_K; ++k) {
    const int j = idx[((size_t)b * NPTS + n) * KNN_K + k];
    __syncthreads();
    for (int c = t; c < C; c += blockDim.x) nbr[c] = x[(size_t)c * NPTS + j];
    __syncthreads();
    float acc = bi;
    for (int c = 0; c < C; ++c)
      acc += w0[c] * (nbr[c] - ctr[c]) + w0[C + c] * ctr[c];
    vmax = fmaxf(vmax, fmaxf(acc * gg + bb, 0.f));
  }
  out[(size_t)b * ldo + (size_t)t * NPTS + n] = vmax;
}

// ---------------------------------------------------------------------------
// Fused t1(6->64) -> t2(64->128) edge convs + max over k. Block per (b,n), 128 thr.
// ---------------------------------------------------------------------------
__global__ __launch_bounds__(128)
void t1t2_max(const float* __restrict__ X, const int* __restrict__ idx,
              const float* __restrict__ W1, const float* __restrict__ b1,
              const float* __restrict__ g1, const float* __restrict__ bt1,
              const float* __restrict__ W2, const float* __restrict__ b2,
              const float* __restrict__ g2, const float* __restrict__ bt2,
              float* __restrict__ out) {
  __shared__ float ctr[3];
  __shared__ float nbr[3];
  __shared__ float t1buf[64];
  const int b = blockIdx.y, n = blockIdx.x, t = threadIdx.x;
  const float* __restrict__ x = X + (size_t)b * 3 * NPTS;
  if (t < 3) ctr[t] = x[(size_t)t * NPTS + n];
  const float bi2 = b2[t];
  const float gg2 = g2[t] * BN_INV;
  const float bb2 = bt2[t];
  const float* __restrict__ w2 = W2 + (size_t)t * 64;
  float vmax = -3.0e38f;
  for (int k = 0; k < KNN_K; ++k) {
    const int j = idx[((size_t)b * NPTS + n) * KNN_K + k];
    __syncthreads();
    if (t < 3) nbr[t] = x[(size_t)t * NPTS + j];
    __syncthreads();
    if (t < 64) {
      float acc = b1[t];
      const float* w1 = W1 + (size_t)t * 6;
      for (int c = 0; c < 3; ++c)
        acc += w1[c] * (nbr[c] - ctr[c]) + w1[3 + c] * ctr[c];
      t1buf[t] = fmaxf(acc * (g1[t] * BN_INV) + bt1[t], 0.f);
    }
    __syncthreads();
    float acc = bi2;
    for (int c = 0; c < 64; ++c) acc += w2[c] * t1buf[c];
    vmax = fmaxf(vmax, fmaxf(acc * gg2 + bb2, 0.f));
  }
  out[(size_t)b * 128 * NPTS + (size_t)t * NPTS + n] = vmax;
}

// trans = reshape(t @ tW + eye(3)); x[b,d,n] = sum_c x0[b,c,n]*trans[b,c,d]
__global__ __launch_bounds__(256)
void trans_apply(const float* __restrict__ t, const float* __restrict__ tW,
                 const float* __restrict__ x0, float* __restrict__ xT) {
  __shared__ float tr[9];
  const int b = blockIdx.x, tid = threadIdx.x;
  if (tid < 9) {
    float acc = (tid == 0 || tid == 4 || tid == 8) ? 1.f : 0.f;
    for (int c = 0; c < 256; ++c) acc += t[(size_t)b * 256 + c] * tW[c * 9 + tid];
    tr[tid] = acc;
  }
  __syncthreads();
  for (int n = tid; n < NPTS; n += 256) {
    const float v0 = x0[(size_t)b * 3072 + n];
    const float v1 = x0[(size_t)b * 3072 + 1024 + n];
    const float v2 = x0[(size_t)b * 3072 + 2048 + n];
    for (int d = 0; d < 3; ++d)
      xT[(size_t)b * 3072 + (size_t)d * NPTS + n] =
          v0 * tr[d] + v1 * tr[3 + d] + v2 * tr[6 + d];
  }
}

__global__ void viewmax_kernel(const float* __restrict__ v, float* __restrict__ o, int n) {
  const int i = blockIdx.x * blockDim.x + threadIdx.x;
  if (i >= n) return;
  const int b = i / 4096, j = i % 4096;
  float m = -3.0e38f;
  for (int k = 0; k < 12; ++k) m = fmaxf(m, v[((size_t)b * 12 + k) * 4096 + j]);
  o[i] = m;
}

__global__ void bn_elem(const float* __restrict__ x, const float* __restrict__ g,
                        const float* __restrict__ bta, float* __restrict__ o,
                        int feat, int n) {
  const int i = blockIdx.x * blockDim.x + threadIdx.x;
  if (i >= n) return;
  const int j = i % feat;
  o[i] = x[i] * (g[j] * BN_INV) + bta[j];
}

__global__ void concat3(const float* __restrict__ a, const float* __restrict__ b,
                        const float* __restrict__ c, float* __restrict__ o, int feat) {
  const int i = blockIdx.x * blockDim.x + threadIdx.x;
  if (i >= 16 * feat) return;
  const int bb = i / feat, j = i % feat;
  o[(size_t)bb * 3 * feat + j] = a[i];
  o[(size_t)bb * 3 * feat + feat + j] = b[i];
  o[(size_t)bb * 3 * feat + 2 * feat + j] = c[i];
}

__global__ __launch_bounds__(256)
void softmax_row(float* __restrict__ x, int ncol) {
  __shared__ float red[256];
  const int b = blockIdx.x, t = threadIdx.x;
  float* row = x + (size_t)b * ncol;
  float m = -3.0e38f;
  for (int j = t; j < ncol; j += 256) m = fmaxf(m, row[j]);
  red[t] = m; __syncthreads();
  for (int s = 128; s > 0; s >>= 1) { if (t < s) red[t] = fmaxf(red[t], red[t + s]); __syncthreads(); }
  m = red[0]; __syncthreads();
  float sum = 0.f;
  for (int j = t; j < ncol; j += 256) { float e = expf(row[j] - m); row[j] = e; sum += e; }
  red[t] = sum; __syncthreads();
  for (int s = 128; s > 0; s >>= 1) { if (t < s) red[t] += red[t + s]; __syncthreads(); }
  const float inv = 1.f / red[0];
  __syncthreads();
  for (int j = t; j < ncol; j += 256) row[j] *= inv;
}

// ---------------------------------------------------------------------------
extern "C" void kernel_launch(void* const* d_in, const int* in_sizes, int n_in,
                              void* d_out, int out_size, void* d_ws, size_t ws_size,
                              hipStream_t stream) {
  (void)in_sizes; (void)n_in; (void)out_size; (void)ws_size;
  const float* pc = (const float*)d_in[0];      // (16,3,1024,1) == (16,3,1024)
  const float* mv = (const float*)d_in[1];      // (16,12,4096)
  const float* dp = (const float*)d_in[2];
#define P(i) ((const float*)d_in[i])
  // param leaf indices (dict flatten order)
  enum {
    T1W=3,T1B,T1G,T1T, T2W,T2B,T2G,T2T, T3W,T3B,T3G,T3T,
    TF1W,TF1B,TF1G,TF1T, TF2W,TF2B,TF2G,TF2T, TWm,
    C1W,C1B,C1G,C1T, C2W,C2B,C2G,C2T, C3W,C3B,C3G,C3T, C4W,C4B,C4G,C4T,
    C5W,C5B,C5G,C5T, FMW,FMB,FMG,FMT,
    G1W,G1B, G2W,G2B, G3W,G3B, G4W,G4B,
    BN2G,BN2T, BN3G,BN3T, BN1G,BN1T,
    D1W,D1B, D2W,D2B, DBNG,DBNT, D3W,D3B,
    DI1W,DI1B, DI2W,DI2B, DI3W,DI3B,
    CBG,CBT, CL1W,CL1B, CL2W,CL2B,
    F1W,F1B,F1G,F1T, F2W,F2B,F2G,F2T, F3W,F3B
  };

  // output offsets (floats)
  float* out = (float*)d_out;
  float* o_pcfeat = out;                 // 16x1024
  float* o_mvf    = out + 16384;
  float* o_dpf    = out + 32768;
  float* o_enc    = out + 49152;         // 3 x 16x512
  float* o_dec    = out + 73728;         // 3 x 16x1024
  float* o_dis    = out + 122880;        // 3 x 16x1
  float* o_cls    = out + 122928;        // 3 x 16x1024
  float* o_net    = out + 172080;        // 16x256
  float* o_pred   = out + 176176;        // 16x40

  // workspace carve-up
  char* wsp = (char*)d_ws;
  auto alloc = [&](size_t bytes) -> void* {
    void* p = (void*)wsp;
    wsp += (bytes + 255) & ~(size_t)255;
    return p;
  };
  int*   w_idx   = (int*)  alloc((size_t)16 * NPTS * KNN_K * 4);
  float* w_tmax  = (float*)alloc((size_t)16 * 128 * NPTS * 4);
  float* w_cat   = (float*)alloc((size_t)16 * 320 * NPTS * 4);   // x1|x2|x3|x4
  float* w_xT    = (float*)alloc((size_t)16 * 3 * NPTS * 4);
  float* w_tfeat = (float*)alloc((size_t)16 * 1024 * 4);
  float* w_tf1   = (float*)alloc((size_t)16 * 512 * 4);
  float* w_tf2   = (float*)alloc((size_t)16 * 256 * 4);
  float* w_vmax  = (float*)alloc((size_t)16 * 4096 * 4);
  float* w_y1    = (float*)alloc((size_t)16 * 10240 * 4);
  float* w_y2    = (float*)alloc((size_t)16 * 5120 * 4);
  float* w_y3    = (float*)alloc((size_t)16 * 512 * 4);
  float* w_z1    = (float*)alloc((size_t)16 * 512 * 4);
  float* w_z2    = (float*)alloc((size_t)16 * 64 * 4);
  float* w_h1    = (float*)alloc((size_t)16 * 512 * 4);
  float* w_h2    = (float*)alloc((size_t)16 * 512 * 4);
  float* w_catf  = (float*)alloc((size_t)16 * 3072 * 4);
  float* w_fo1   = (float*)alloc((size_t)16 * 1024 * 4);

  const dim3 gridBN(NPTS, 16);
  // ---- T-Net ----
  knn_kernel<<<gridBN, 256, 0, stream>>>(pc, w_idx, 3, 3 * NPTS);
  t1t2_max<<<gridBN, 128, 0, stream>>>(pc, w_idx, P(T1W), P(T1B), P(T1G), P(T1T),
                                       P(T2W), P(T2B), P(T2G), P(T2T), w_tmax);
  convmax_wmma<<<dim3(64, 16), 32, 0, stream>>>(w_tmax, P(T3W), P(T3B), P(T3G), P(T3T),
                                                w_tfeat, 128, NPTS, 1024, 128 * NPTS);
  gemm16_wmma<<<32, 32, 0, stream>>>(w_tfeat, P(TF1W), P(TF1B), P(TF1G), P(TF1T),
                                     w_tf1, 1024, 512, 1, 0.f);
  gemm16_wmma<<<16, 32, 0, stream>>>(w_tf1, P(TF2W), P(TF2B), P(TF2G), P(TF2T),
                                     w_tf2, 512, 256, 1, 0.f);
  trans_apply<<<16, 256, 0, stream>>>(w_tf2, P(TWm), pc, w_xT);

  // ---- DGCNN trunk: x1..x4 written into channel slices of w_cat ----
  const int ldcat = 320 * NPTS;
  knn_kernel<<<gridBN, 256, 0, stream>>>(w_xT, w_idx, 3, 3 * NPTS);
  edgeconv_max<<<gridBN, 64, 0, stream>>>(w_xT, w_idx, P(C1W), P(C1B), P(C1G), P(C1T),
                                          w_cat + 0 * NPTS, 3, 64, 3 * NPTS, ldcat);
  knn_kernel<<<gridBN, 256, 0, stream>>>(w_cat + 0 * NPTS, w_idx, 64, ldcat);
  edgeconv_max<<<gridBN, 64, 0, stream>>>(w_cat + 0 * NPTS, w_idx, P(C2W), P(C2B), P(C2G), P(C2T),
                                          w_cat + 64 * NPTS, 64, 64, ldcat, ldcat);
  knn_kernel<<<gridBN, 256, 0, stream>>>(w_cat + 64 * NPTS, w_idx, 64, ldcat);
  edgeconv_max<<<gridBN, 64, 0, stream>>>(w_cat + 64 * NPTS, w_idx, P(C3W), P(C3B), P(C3G), P(C3T),
                                          w_cat + 128 * NPTS, 64, 64, ldcat, ldcat);
  knn_kernel<<<gridBN, 256, 0, stream>>>(w_cat + 128 * NPTS, w_idx, 64, ldcat);
  edgeconv_max<<<gridBN, 128, 0, stream>>>(w_cat + 128 * NPTS, w_idx, P(C4W), P(C4B), P(C4G), P(C4T),
                                           w_cat + 192 * NPTS, 64, 128, ldcat, ldcat);
  convmax_wmma<<<dim3(64, 16), 32, 0, stream>>>(w_cat, P(C5W), P(C5B), P(C5G), P(C5T),
                                                o_pcfeat, 320, NPTS, 1024, ldcat);

  // ---- view branches ----
  viewmax_kernel<<<(16 * 4096 + 255) / 256, 256, 0, stream>>>(mv, w_vmax, 16 * 4096);
  gemm16_wmma<<<64, 32, 0, stream>>>(w_vmax, P(FMW), P(FMB), P(FMG), P(FMT),
                                     o_mvf, 4096, 1024, 1, 0.f);
  viewmax_kernel<<<(16 * 4096 + 255) / 256, 256, 0, stream>>>(dp, w_vmax, 16 * 4096);
  gemm16_wmma<<<64, 32, 0, stream>>>(w_vmax, P(FMW), P(FMB), P(FMG), P(FMT),
                                     o_dpf, 4096, 1024, 1, 0.f);

  // ---- encoder / decoder / discriminator / classifier for each modality ----
  const float* feats[3] = { o_pcfeat, o_mvf, o_dpf };
  for (int i = 0; i < 3; ++i) {
    float* enc_i = o_enc + (size_t)i * 16 * 512;
    float* dec_i = o_dec + (size_t)i * 16 * 1024;
    float* dis_i = o_dis + (size_t)i * 16 * 1;
    float* cls_i = o_cls + (size_t)i * 16 * 1024;
    // gen_en: lrelu(bn(linear)) x3, tanh(linear)
    gemm16_wmma<<<640, 32, 0, stream>>>(feats[i], P(G1W), P(G1B), P(BN2G), P(BN2T),
                                        w_y1, 1024, 10240, 2, 0.01f);
    gemm16_wmma<<<320, 32, 0, stream>>>(w_y1, P(G2W), P(G2B), P(BN3G), P(BN3T),
                                        w_y2, 10240, 5120, 2, 0.01f);
    gemm16_wmma<<<32, 32, 0, stream>>>(w_y2, P(G3W), P(G3B), P(BN1G), P(BN1T),
                                       w_y3, 5120, 512, 2, 0.01f);
    gemm16_wmma<<<32, 32, 0, stream>>>(w_y3, P(G4W), P(G4B), nullptr, nullptr,
                                       enc_i, 512, 512, 3, 0.f);
    // gen_dec: lrelu(.2), lrelu(.2, bn), tanh
    gemm16_wmma<<<32, 32, 0, stream>>>(enc_i, P(D1W), P(D1B), nullptr, nullptr,
                                       w_z1, 512, 512, 2, 0.2f);
    gemm16_wmma<<<4, 32, 0, stream>>>(w_z1, P(D2W), P(D2B), P(DBNG), P(DBNT),
                                      w_z2, 512, 64, 2, 0.2f);
    gemm16_wmma<<<64, 32, 0, stream>>>(w_z2, P(D3W), P(D3B), nullptr, nullptr,
                                       dec_i, 64, 1024, 3, 0.f);
    // dis: lrelu(.2), tanh, sigmoid
    gemm16_wmma<<<4, 32, 0, stream>>>(dec_i, P(DI1W), P(DI1B), nullptr, nullptr,
                                      w_z2, 1024, 64, 2, 0.2f);
    gemm16_wmma<<<1, 32, 0, stream>>>(w_z2, P(DI2W), P(DI2B), nullptr, nullptr,
                                      w_h1, 64, 16, 3, 0.f);
    gemm16_wmma<<<1, 32, 0, stream>>>(w_h1, P(DI3W), P(DI3B), nullptr, nullptr,
                                      dis_i, 16, 1, 4, 0.f);
    // cls: bn(e) -> tanh(linear) -> softmax(linear)
    bn_elem<<<(16 * 512 + 255) / 256, 256, 0, stream>>>(enc_i, P(CBG), P(CBT),
                                                        w_h1, 512, 16 * 512);
    gemm16_wmma<<<32, 32, 0, stream>>>(w_h1, P(CL1W), P(CL1B), nullptr, nullptr,
                                       w_h2, 512, 512, 3, 0.f);
    gemm16_wmma<<<64, 32, 0, stream>>>(w_h2, P(CL2W), P(CL2B), nullptr, nullptr,
                                       cls_i, 512, 1024, 0, 0.f);
    softmax_row<<<16, 256, 0, stream>>>(cls_i, 1024);
  }

  // ---- fusion head ----
  concat3<<<(16 * 1024 + 255) / 256, 256, 0, stream>>>(o_pcfeat, o_mvf, o_dpf,
                                                       w_catf, 1024);
  gemm16_wmma<<<64, 32, 0, stream>>>(w_catf, P(F1W), P(F1B), P(F1G), P(F1T),
                                     w_fo1, 3072, 1024, 1, 0.f);
  gemm16_wmma<<<16, 32, 0, stream>>>(w_fo1, P(F2W), P(F2B), P(F2G), P(F2T),
                                     o_net, 1024, 256, 1, 0.f);
  gemm16_wmma<<<3, 32, 0, stream>>>(o_net, P(F3W), P(F3B), nullptr, nullptr,
                                    o_pred, 256, 40, 0, 0.f);
#undef P
}